// GNNExtrapolation_63041529970790
// MI455X (gfx1250) — compile-verified
//
#include <hip/hip_runtime.h>

typedef __attribute__((ext_vector_type(2))) float v2f;
typedef __attribute__((ext_vector_type(8))) float v8f;

// Problem constants (from reference)
#define PB    16      // batch
#define PT    12      // T_IN
#define PN    5000    // nodes
#define PC    4       // channels
#define PK    16      // neighbors
#define PH    4       // heads
#define DIN   48      // T*H
#define DOUT  12      // T_TOTAL - T_IN
#define TOUT  24      // T_TOTAL

// ---------------------------------------------------------------------------
// Kernel 1: copy x -> out[:, 0:12, :, :]  (contiguous per-batch block copy)
// 960000 float4 total: per b, 60000 float4 src -> offset b*120000 dst
// ---------------------------------------------------------------------------
__global__ __launch_bounds__(256) void gnn_copy_kernel(
    const float4* __restrict__ x, float4* __restrict__ out) {
  const int i = blockIdx.x * blockDim.x + threadIdx.x;     // [0, 960000)
  const int b = i / 60000;                                  // PT*PN*PC/4
  const int r = i - b * 60000;
  out[(size_t)b * 120000 + r] = x[i];                       // TOUT*PN*PC/4
}

// ---------------------------------------------------------------------------
// Kernel 2: fused gather + exp-weight aggregation + WMMA GEMM + ReLU
// One wave handles a 16-row tile (4 nodes x 4 channels) of one batch.
//   Rows  M=0..15  -> node n0 + M/4, channel M%4
//   Lane half hi   -> h pair {2hi, 2hi+1}  == A-fragment K split for 16x16x4
//   N dim (cols)   -> output o (12 valid, padded to 16)
// ---------------------------------------------------------------------------
__global__ __launch_bounds__(256) void gnn_wmma_kernel(
    const float* __restrict__ x, const float* __restrict__ dists,
    const float* __restrict__ W, const float* __restrict__ bias,
    const int* __restrict__ neighbors, float* __restrict__ out) {
  const int lane = threadIdx.x & 31;
  const int wave = blockIdx.x * (blockDim.x >> 5) + (threadIdx.x >> 5);

  // 20000 tiles: 1250 per batch, each tile = 4 consecutive nodes
  const int b  = wave / 1250;
  const int n0 = (wave - b * 1250) * 4;

  const int M  = lane & 15;        // row within 16-row tile
  const int hi = lane >> 4;        // 0 -> h={0,1}, 1 -> h={2,3}
  const int n  = n0 + (M >> 2);    // node for this lane's row
  const int c  = M & 3;            // channel for this lane's row

  // lam[h] = (h+1)/H ; this lane's two heads
  const float lam0 = (float)(2 * hi + 1) * 0.25f;
  const float lam1 = (float)(2 * hi + 2) * 0.25f;
  const float inv_s2 = 1.0f / 36.0f;   // 1/SIGMA^2

  // Per-lane accumulators == A fragments: acc0[t] = feat[row][4t+2hi],
  // acc1[t] = feat[row][4t+2hi+1]  (exactly the 16x16x4 f32 A layout)
  float acc0[PT], acc1[PT];
#pragma unroll
  for (int t = 0; t < PT; ++t) { acc0[t] = 0.f; acc1[t] = 0.f; }

  const size_t xbase = (size_t)b * PT * PN * PC + c;
#pragma unroll 4
  for (int k = 0; k < PK; ++k) {
    const int   nbr = neighbors[n * PK + k];
    const float d   = dists[n * PK + k];
    const float d2  = d * d * inv_s2;
    const float w0  = __expf(-d2 * lam0);
    const float w1  = __expf(-d2 * lam1);
    const float* xp = x + xbase + (size_t)nbr * PC;
#pragma unroll
    for (int t = 0; t < PT; ++t) {
      const float xv = xp[(size_t)t * (PN * PC)];
      acc0[t] = fmaf(xv, w0, acc0[t]);
      acc1[t] = fmaf(xv, w1, acc1[t]);
    }
  }

  // B fragments: W is (48 x 12) row-major; B layout for 16x16x4 f32:
  //   VGPR0: lane<16 -> K row 4s+0 ; lane>=16 -> K row 4s+2   (we use 2hi)
  //   VGPR1: lane<16 -> K row 4s+1 ; lane>=16 -> K row 4s+3
  const int  o      = lane & 15;
  const bool ovalid = (o < DOUT);
  v2f bfrag[PT];
#pragma unroll
  for (int s = 0; s < PT; ++s) {
    const int k0 = 4 * s + 2 * hi;
    bfrag[s].x = ovalid ? W[k0 * DOUT + o] : 0.f;
    bfrag[s].y = ovalid ? W[(k0 + 1) * DOUT + o] : 0.f;
  }

  // C init = bias[o] broadcast over M (C layout: N = lane%16 everywhere)
  v8f cacc;
  const float bo = ovalid ? bias[o] : 0.f;
#pragma unroll
  for (int j = 0; j < 8; ++j) cacc[j] = bo;

  // 12 chained WMMAs cover K = 48
#pragma unroll
  for (int s = 0; s < PT; ++s) {
    v2f a;
    a.x = acc0[s];
    a.y = acc1[s];
    cacc = __builtin_amdgcn_wmma_f32_16x16x4_f32(
        /*neg_a=*/false, a, /*neg_b=*/false, bfrag[s],
        /*c_mod=*/(short)0, cacc, /*reuse_a=*/false, /*reuse_b=*/false);
  }

  // ReLU + scatter: D VGPR j -> (M = j + 8*hi, N = o); out[b, 12+o, n, c]
  if (ovalid) {
#pragma unroll
    for (int j = 0; j < 8; ++j) {
      const int M2 = j + 8 * hi;
      const int nn = n0 + (M2 >> 2);
      const int cc = M2 & 3;
      const float v = fmaxf(cacc[j], 0.f);
      out[((size_t)b * TOUT + PT + o) * (PN * PC) + (size_t)nn * PC + cc] = v;
    }
  }
}

// ---------------------------------------------------------------------------
extern "C" void kernel_launch(void* const* d_in, const int* in_sizes, int n_in,
                              void* d_out, int out_size, void* d_ws, size_t ws_size,
                              hipStream_t stream) {
  const float* x     = (const float*)d_in[0];
  const float* dists = (const float*)d_in[1];
  const float* W     = (const float*)d_in[2];
  const float* bias  = (const float*)d_in[3];
  const int* neigh   = (const int*)d_in[4];
  float* out = (float*)d_out;

  // Copy x into first 12 time slices: 960000 float4s
  gnn_copy_kernel<<<3750, 256, 0, stream>>>((const float4*)x, (float4*)out);

  // 20000 wave-tiles, 8 waves per 256-thread block -> 2500 blocks
  gnn_wmma_kernel<<<2500, 256, 0, stream>>>(x, dists, W, bias, neigh, out);
}